// GCN_19722489823527
// MI455X (gfx1250) — compile-verified
//
#include <hip/hip_runtime.h>
#include <math.h>

typedef __attribute__((ext_vector_type(2))) float v2f;
typedef __attribute__((ext_vector_type(8))) float v8f;

#define D_IN  128
#define H_DIM 32
#define C_DIM 5

// ---------- weight repacking ----------
// W1cat [128 x 64] = [W_rel1 | W_root1]
__global__ void gcn_build_w1(const float* __restrict__ Wrel, const float* __restrict__ Wroot,
                             float* __restrict__ Wcat) {
    int i = blockIdx.x * 256 + threadIdx.x;
    if (i >= 128 * 64) return;
    int k = i >> 6, j = i & 63;
    Wcat[i] = (j < 32) ? Wrel[k * 32 + j] : Wroot[k * 32 + (j - 32)];
}

// W2p [32 x 16] = [W_rel2 | W_root2 | zero-pad]
__global__ void gcn_build_w2(const float* __restrict__ Wrel, const float* __restrict__ Wroot,
                             float* __restrict__ Wp) {
    int i = blockIdx.x * 256 + threadIdx.x;
    if (i >= 32 * 16) return;
    int k = i >> 4, n = i & 15;
    float v = 0.0f;
    if (n < 5)       v = Wrel[k * 5 + n];
    else if (n < 10) v = Wroot[k * 5 + (n - 5)];
    Wp[i] = v;
}

__global__ void gcn_zero(float* __restrict__ p, long n) {
    long i = (long)blockIdx.x * 256 + threadIdx.x;
    if (i < n) p[i] = 0.0f;
}

// ---------- GEMM1: PR1[N,64] = x[N,128] @ W1cat[128,64], f32 WMMA 16x16x4 ----------
__global__ __launch_bounds__(256)
void gcn_gemm1_wmma(const float* __restrict__ x, const float* __restrict__ W,
                    float* __restrict__ PR1, int nTiles) {
    int wave = threadIdx.x >> 5;
    int lane = threadIdx.x & 31;
    int tile = blockIdx.x * 8 + wave;
    if (tile >= nTiles) return;
    int half = lane >> 4;      // 0: K pair {0,1}; 1: K pair {2,3} within a k-step
    int m    = lane & 15;
    const float* xr = x + (long)(tile * 16 + m) * D_IN;

    // preload all A fragments for this 16-row tile (K=128 -> 32 steps of K=4)
    v2f a[32];
#pragma unroll
    for (int ks = 0; ks < 32; ++ks) {
        int k = ks * 4 + half * 2;
        a[ks].x = xr[k];
        a[ks].y = xr[k + 1];
    }

#pragma unroll
    for (int ct = 0; ct < 4; ++ct) {          // 4 column tiles of 16 -> 64 outputs
        int n = ct * 16 + m;
        v8f acc = {};
#pragma unroll
        for (int ks = 0; ks < 32; ++ks) {
            int k = ks * 4 + half * 2;
            v2f b;
            b.x = W[k * 64 + n];
            b.y = W[(k + 1) * 64 + n];
            acc = __builtin_amdgcn_wmma_f32_16x16x4_f32(false, a[ks], false, b,
                                                        (short)0, acc, false, false);
        }
#pragma unroll
        for (int r = 0; r < 8; ++r) {
            int orow = tile * 16 + r + half * 8;
            PR1[(long)orow * 64 + ct * 16 + m] = acc[r];
        }
    }
}

// ---------- GEMM2: PR2[N,10] = h[N,32] @ W2p[32,16] (cols 10..15 are pad) ----------
__global__ __launch_bounds__(256)
void gcn_gemm2_wmma(const float* __restrict__ h, const float* __restrict__ W,
                    float* __restrict__ PR2, int nTiles) {
    int wave = threadIdx.x >> 5;
    int lane = threadIdx.x & 31;
    int tile = blockIdx.x * 8 + wave;
    if (tile >= nTiles) return;
    int half = lane >> 4;
    int m    = lane & 15;
    const float* hr = h + (long)(tile * 16 + m) * H_DIM;

    v2f a[8];
#pragma unroll
    for (int ks = 0; ks < 8; ++ks) {
        int k = ks * 4 + half * 2;
        a[ks].x = hr[k];
        a[ks].y = hr[k + 1];
    }
    v8f acc = {};
#pragma unroll
    for (int ks = 0; ks < 8; ++ks) {
        int k = ks * 4 + half * 2;
        v2f b;
        b.x = W[k * 16 + m];
        b.y = W[(k + 1) * 16 + m];
        acc = __builtin_amdgcn_wmma_f32_16x16x4_f32(false, a[ks], false, b,
                                                    (short)0, acc, false, false);
    }
    if (m < 10) {
#pragma unroll
        for (int r = 0; r < 8; ++r) {
            int orow = tile * 16 + r + half * 8;
            PR2[(long)orow * 10 + m] = acc[r];
        }
    }
}

// ---------- edge aggregation, layer 1: one wave per edge, 32 features ----------
__global__ __launch_bounds__(256)
void gcn_agg1(const int* __restrict__ src, const int* __restrict__ dst,
              const float* __restrict__ ew, const float* __restrict__ PR1,
              float* __restrict__ S1, float* __restrict__ cnt, int E) {
    int e = blockIdx.x * 8 + (threadIdx.x >> 5);
    int lane = threadIdx.x & 31;
    if (e >= E) return;
    int s = src[e], d = dst[e];
    float v = ew[e] * PR1[(long)s * 64 + lane];     // cols 0..31 = rel-projection
    atomicAdd(&S1[(long)d * 32 + lane], v);
    if (lane == 0) atomicAdd(&cnt[d], 1.0f);
}

// h = relu(S1/max(cnt,1) + root_proj + b1)
__global__ __launch_bounds__(256)
void gcn_combine1(const float* __restrict__ S1, const float* __restrict__ PR1,
                  const float* __restrict__ b1, const float* __restrict__ cnt,
                  float* __restrict__ h, long total) {
    long i = (long)blockIdx.x * 256 + threadIdx.x;
    if (i >= total) return;
    long n = i >> 5;
    int  j = (int)(i & 31);
    float c = fmaxf(cnt[n], 1.0f);
    float v = S1[i] / c + PR1[n * 64 + 32 + j] + b1[j];
    h[i] = fmaxf(v, 0.0f);
}

// ---------- edge aggregation, layer 2: 8 lanes per edge, 5 features ----------
__global__ __launch_bounds__(256)
void gcn_agg2(const int* __restrict__ src, const int* __restrict__ dst,
              const float* __restrict__ PR2, float* __restrict__ S2, int E) {
    long t = (long)blockIdx.x * 256 + threadIdx.x;
    int e = (int)(t >> 3);
    int c = (int)(t & 7);
    if (e >= E || c >= 5) return;
    atomicAdd(&S2[(long)dst[e] * 5 + c], PR2[(long)src[e] * 10 + c]);
}

// out = log_softmax(S2/max(cnt,1) + root_proj2 + b2)
__global__ __launch_bounds__(256)
void gcn_output(const float* __restrict__ S2, const float* __restrict__ PR2,
                const float* __restrict__ b2, const float* __restrict__ cnt,
                float* __restrict__ out, int N) {
    int n = blockIdx.x * 256 + threadIdx.x;
    if (n >= N) return;
    float c = fmaxf(cnt[n], 1.0f);
    float v[5];
#pragma unroll
    for (int j = 0; j < 5; ++j)
        v[j] = S2[(long)n * 5 + j] / c + PR2[(long)n * 10 + 5 + j] + b2[j];
    float mx = v[0];
#pragma unroll
    for (int j = 1; j < 5; ++j) mx = fmaxf(mx, v[j]);
    float s = 0.0f;
#pragma unroll
    for (int j = 0; j < 5; ++j) s += expf(v[j] - mx);
    float l = logf(s);
#pragma unroll
    for (int j = 0; j < 5; ++j) out[(long)n * 5 + j] = v[j] - mx - l;
}

extern "C" void kernel_launch(void* const* d_in, const int* in_sizes, int n_in,
                              void* d_out, int out_size, void* d_ws, size_t ws_size,
                              hipStream_t stream) {
    const float* x      = (const float*)d_in[0];
    const int*   ei     = (const int*)d_in[1];
    const float* ew     = (const float*)d_in[2];
    const float* Wrel1  = (const float*)d_in[3];
    const float* Wroot1 = (const float*)d_in[4];
    const float* b1     = (const float*)d_in[5];
    const float* Wrel2  = (const float*)d_in[6];
    const float* Wroot2 = (const float*)d_in[7];
    const float* b2     = (const float*)d_in[8];

    const int N = in_sizes[0] / D_IN;      // 100000
    const int E = in_sizes[1] / 2;         // 1600000
    const int* src = ei;
    const int* dst = ei + E;

    float* ws = (float*)d_ws;
    long off = 0;
    float* PR1 = ws + off; off += (long)N * 64;   // [rel_proj | root_proj] layer1
    float* S1  = ws + off; off += (long)N * 32;   // layer1 weighted sums
    float* cnt = ws + off; off += N;              // in-degree (shared by both layers)
    float* h   = ws + off; off += (long)N * 32;   // hidden activations
    float* W1c = ws + off; off += 128 * 64;
    float* W2p = ws + off; off += 32 * 16;
    float* PR2 = PR1;   // reuse: PR1 dead after combine1
    float* S2  = S1;    // reuse: S1 dead after combine1

    const int nTiles = N / 16;   // N % 16 == 0

    gcn_build_w1<<<(128 * 64 + 255) / 256, 256, 0, stream>>>(Wrel1, Wroot1, W1c);
    gcn_build_w2<<<(32 * 16 + 255) / 256, 256, 0, stream>>>(Wrel2, Wroot2, W2p);

    gcn_gemm1_wmma<<<(nTiles + 7) / 8, 256, 0, stream>>>(x, W1c, PR1, nTiles);

    long nz1 = (long)N * 33;   // S1 and cnt are contiguous
    gcn_zero<<<(int)((nz1 + 255) / 256), 256, 0, stream>>>(S1, nz1);
    gcn_agg1<<<(E + 7) / 8, 256, 0, stream>>>(src, dst, ew, PR1, S1, cnt, E);

    long nh = (long)N * 32;
    gcn_combine1<<<(int)((nh + 255) / 256), 256, 0, stream>>>(S1, PR1, b1, cnt, h, nh);

    gcn_gemm2_wmma<<<(nTiles + 7) / 8, 256, 0, stream>>>(h, W2p, PR2, nTiles);

    long nz2 = (long)N * 5;
    gcn_zero<<<(int)((nz2 + 255) / 256), 256, 0, stream>>>(S2, nz2);
    gcn_agg2<<<(int)(((long)E * 8 + 255) / 256), 256, 0, stream>>>(src, dst, PR2, S2, E);

    gcn_output<<<(N + 255) / 256, 256, 0, stream>>>(S2, PR2, b2, cnt, (float*)d_out, N);
}